// SSIR_ODEFIT_87935160418917
// MI455X (gfx1250) — compile-verified
//
#include <hip/hip_runtime.h>
#include <math.h>

// ---------------------------------------------------------------------------
// Problem constants (from reference): B=4, T=96, N=512, row length = 2+N = 514
// Roofline: 404MB params read + 404MB epiparams write + 384MB cs re-read
// => ~1.2GB @ 23.3 TB/s => ~51us floor. Compute is negligible; stay in f32.
// ---------------------------------------------------------------------------
#define EPI_EPS        1e-8f
#define NNODES         512
#define ROWLEN         514
#define NBATCH         4
#define TSTEPS         96
#define WGS_PER_BATCH  16
#define ROWS_PER_WG    32              // 16 WGs * 32 rows = 512 rows per batch
#define OUTVIEW_FLOATS (NBATCH * TSTEPS * NNODES * 3)   // 589824

typedef __attribute__((ext_vector_type(2))) float v2f;
typedef __attribute__((ext_vector_type(8))) float v8f;

// ---------------------------------------------------------------------------
// Kernel A: epiparams = [sigmoid(p0), sigmoid(p1), softmax(p[2:514])]
// One wave32 per row; 16 elements/lane, coalesced 128B accesses; shuffle
// reductions for max and sum. Bandwidth-bound streaming kernel.
// ---------------------------------------------------------------------------
__global__ void __launch_bounds__(256) epi_params_kernel(
    const float* __restrict__ params, float* __restrict__ epi, int nrows) {
  const int wave = threadIdx.x >> 5;
  const int lane = threadIdx.x & 31;
  const long long row = (long long)blockIdx.x * 8 + wave;
  if (row >= nrows) return;

  const float* p = params + (size_t)row * ROWLEN;
  float*       o = epi    + (size_t)row * ROWLEN;

  float v[16];
  float m = -INFINITY;
#pragma unroll
  for (int j = 0; j < 16; ++j) {
    v[j] = p[2 + lane + 32 * j];              // coalesced 128B per access
    m = fmaxf(m, v[j]);
  }
#pragma unroll
  for (int off = 16; off > 0; off >>= 1)
    m = fmaxf(m, __shfl_xor(m, off, 32));     // wave32 max-reduce

  float s = 0.f;
#pragma unroll
  for (int j = 0; j < 16; ++j) { v[j] = expf(v[j] - m); s += v[j]; }
#pragma unroll
  for (int off = 16; off > 0; off >>= 1)
    s += __shfl_xor(s, off, 32);              // wave32 sum-reduce

  const float inv = 1.0f / s;
#pragma unroll
  for (int j = 0; j < 16; ++j)
    o[2 + lane + 32 * j] = v[j] * inv;

  if (lane < 2) {                              // beta, gamma sigmoids
    const float x = p[lane];
    o[lane] = 1.0f / (1.0f + expf(-x));
  }
}

// ---------------------------------------------------------------------------
// Per-batch sense barrier: fresh counter per (batch, step), zeroed by a
// hipMemsetAsync each launch (graph-capture safe). Release via threadfence
// before arrive; acquire load in the poll; s_sleep backoff while waiting.
// ---------------------------------------------------------------------------
__device__ __forceinline__ void batch_barrier(int* cnt) {
  __threadfence();
  __syncthreads();
  if (threadIdx.x == 0) {
    atomicAdd(cnt, 1);
    while (__hip_atomic_load(cnt, __ATOMIC_ACQUIRE, __HIP_MEMORY_SCOPE_AGENT)
           < WGS_PER_BATCH)
      __builtin_amdgcn_s_sleep(8);
  }
  __syncthreads();
}

// ---------------------------------------------------------------------------
// Kernel B: the 96-step SIR scan. 16 persistent WGs per batch (64 total),
// 256 threads = 8 waves each; WG owns 32 rows. Per step each wave computes a
// 16-row x 128-K slice of infection = c @ I via V_WMMA_F32_16X16X4_F32:
//   A fragment (documented 32-bit 16x4 layout): lane L<16 -> row M=L, K pair
//   {k0,k0+1}; lane L>=16 -> row M=L-16, K pair {k0+2,k0+3}. One 8B load/lane.
//   B fragment: I broadcast to all 16 columns -> every D column = A @ I_chunk,
//   so the fragment value depends only on K: same {k0,k0+1}/{k0+2,k0+3} pick
//   from the LDS-resident I vector (8B broadcast ds loads).
// 4 waves K-split each tile; partials merged with ds_add_f32. Cross-WG I
// exchange via ping-pong global buffer + per-batch barrier (1 per step).
// ---------------------------------------------------------------------------
__global__ void __launch_bounds__(256) sir_scan_kernel(
    const float* __restrict__ x0, const float* __restrict__ epi,
    float* __restrict__ out, float* __restrict__ ibuf,
    int* __restrict__ counters) {
  __shared__ float ldsI[NNODES];
  __shared__ float ldsInf[ROWS_PER_WG];
  __shared__ float ldsS[ROWS_PER_WG];
  __shared__ float ldsR[ROWS_PER_WG];

  const int b    = blockIdx.x / WGS_PER_BATCH;
  const int g    = blockIdx.x % WGS_PER_BATCH;
  const int tid  = threadIdx.x;
  const int lane = tid & 31;
  const int wave = tid >> 5;
  const int tile  = wave >> 2;                 // 0..1: which 16-row tile
  const int kpart = wave & 3;                  // 0..3: 128-wide K slice
  const int row0  = g * ROWS_PER_WG;           // batch-local first row

  // Init: load x0 state; publish initial I into ping buffer 0.
  if (tid < ROWS_PER_WG) {
    const int n = row0 + tid;
    const float* xp = x0 + ((size_t)b * NNODES + n) * 3;
    ldsS[tid] = xp[0];
    ldsR[tid] = xp[2];
    ibuf[(size_t)b * NNODES + n] = xp[1];
  }

  const int mRow   = row0 + tile * 16 + (lane & 15);  // row this lane feeds
  const int kShift = (lane >> 4) << 1;                // 0 or 2 (A-layout halves)

  for (int t = 0; t < TSTEPS; ++t) {
    batch_barrier(counters + b * TSTEPS + t);  // I[t] fully published

    const float* icur = ibuf + (size_t)(t & 1) * (NBATCH * NNODES)
                             + (size_t)b * NNODES;
    ldsI[tid]       = icur[tid];
    ldsI[tid + 256] = icur[tid + 256];
    if (tid < ROWS_PER_WG) ldsInf[tid] = 0.f;
    __syncthreads();

    const size_t rowbase = (size_t)(b * TSTEPS + t) * NNODES;
    const float* crow = epi + (rowbase + mRow) * ROWLEN + 2;  // c[mRow][*]

    int k = kpart * 128 + kShift;
    v8f acc = {0.f, 0.f, 0.f, 0.f, 0.f, 0.f, 0.f, 0.f};
#pragma unroll 8
    for (int it = 0; it < 32; ++it) {
      const v2f a  = *(const v2f*)(crow + k);   // A: c tile, 8B/lane
      const v2f bI = *(const v2f*)(ldsI + k);   // B: I broadcast (LDS)
      acc = __builtin_amdgcn_wmma_f32_16x16x4_f32(
          /*neg_a=*/false, a, /*neg_b=*/false, bI,
          /*c_mod=*/(short)0, acc, /*reuse_a=*/false, /*reuse_b=*/false);
      k += 4;
    }

    // D columns are identical; lane 0 holds rows 0..7, lane 16 rows 8..15.
    if (lane == 0) {
      atomicAdd(&ldsInf[tile * 16 + 0], acc[0]);
      atomicAdd(&ldsInf[tile * 16 + 1], acc[1]);
      atomicAdd(&ldsInf[tile * 16 + 2], acc[2]);
      atomicAdd(&ldsInf[tile * 16 + 3], acc[3]);
      atomicAdd(&ldsInf[tile * 16 + 4], acc[4]);
      atomicAdd(&ldsInf[tile * 16 + 5], acc[5]);
      atomicAdd(&ldsInf[tile * 16 + 6], acc[6]);
      atomicAdd(&ldsInf[tile * 16 + 7], acc[7]);
    } else if (lane == 16) {
      atomicAdd(&ldsInf[tile * 16 +  8], acc[0]);
      atomicAdd(&ldsInf[tile * 16 +  9], acc[1]);
      atomicAdd(&ldsInf[tile * 16 + 10], acc[2]);
      atomicAdd(&ldsInf[tile * 16 + 11], acc[3]);
      atomicAdd(&ldsInf[tile * 16 + 12], acc[4]);
      atomicAdd(&ldsInf[tile * 16 + 13], acc[5]);
      atomicAdd(&ldsInf[tile * 16 + 14], acc[6]);
      atomicAdd(&ldsInf[tile * 16 + 15], acc[7]);
    }
    __syncthreads();

    // State update for this WG's 32 rows (matches reference step()).
    if (tid < ROWS_PER_WG) {
      const int n = row0 + tid;
      const float S   = ldsS[tid];
      const float R   = ldsR[tid];
      const float I   = ldsI[n];
      const float inf = ldsInf[tid];
      const float* bg = epi + (rowbase + n) * ROWLEN;  // sigmoided by kernel A
      const float beta  = bg[0];
      const float gamma = bg[1];
      const float Np = fmaxf(S + I + R, EPI_EPS);
      const float dS = -beta * S / Np * inf;
      const float dR = gamma * I;
      const float dI = -dS - dR;
      float St = fmaxf(S + dS, 0.f);
      float It = fmaxf(I + dI, 0.f);
      float Rt = fmaxf(R + dR, 0.f);
      const float scale = Np / fmaxf(St + It + Rt, EPI_EPS);
      St *= scale; It *= scale; Rt *= scale;
      ldsS[tid] = St;
      ldsR[tid] = Rt;
      ibuf[(size_t)((t + 1) & 1) * (NBATCH * NNODES) + (size_t)b * NNODES + n] = It;
      float* op = out + (rowbase + n) * 3;
      op[0] = St; op[1] = It; op[2] = Rt;
    }
    // Next iteration's batch_barrier (threadfence + syncthreads) orders the
    // ldsI/ldsS rewrites and the cross-WG ibuf exchange.
  }
}

// ---------------------------------------------------------------------------
// Launch: memset barrier counters (graph-safe), then kernel A -> kernel B
// (stream-ordered: B consumes epiparams that A wrote into d_out).
// d_ws layout: [0,16KB) ping-pong I buffers, [16KB, 16KB+1.5KB) counters.
// ---------------------------------------------------------------------------
extern "C" void kernel_launch(void* const* d_in, const int* in_sizes, int n_in,
                              void* d_out, int out_size, void* d_ws, size_t ws_size,
                              hipStream_t stream) {
  (void)in_sizes; (void)n_in; (void)out_size; (void)ws_size;
  const float* x0     = (const float*)d_in[0];
  const float* params = (const float*)d_in[1];
  float* out = (float*)d_out;
  float* epi = out + OUTVIEW_FLOATS;           // epiparams section of d_out

  float* ibuf     = (float*)d_ws;              // 2 * 4 * 512 floats
  int*   counters = (int*)((char*)d_ws + 2 * NBATCH * NNODES * sizeof(float));

  hipMemsetAsync(counters, 0, NBATCH * TSTEPS * sizeof(int), stream);

  const int nrows = NBATCH * TSTEPS * NNODES;  // 196608
  epi_params_kernel<<<nrows / 8, 256, 0, stream>>>(params, epi, nrows);
  sir_scan_kernel<<<NBATCH * WGS_PER_BATCH, 256, 0, stream>>>(
      x0, epi, out, ibuf, counters);
}